// ZeroInflationLossBase_17970143167413
// MI455X (gfx1250) — compile-verified
//
#include <hip/hip_runtime.h>
#include <hip/hip_bf16.h>
#include <cmath>

#define B_ROWS 4096
#define N_COLS 8000
#define N4 2000                 // N_COLS / 4 float4 columns
#define TPB 256
#define COL_BLOCKS 8            // ceil(N4 / TPB)
#define ROW_BLOCKS 128
#define ROWS_PER_BLOCK 32       // B_ROWS / ROW_BLOCKS
#define NPART (COL_BLOCKS * ROW_BLOCKS)   // 1024 partial sums
#define LOG2PI_F 1.8378770664093453f
#define PF_DIST 8
#define INF_BITS 0x7F800000u

typedef float v2f __attribute__((ext_vector_type(2)));
typedef float v8f __attribute__((ext_vector_type(8)));

// ---- exact f32 wave32 sum using V_WMMA_F32_16X16X4_F32 with all-ones B ----
// A layout (16x4 f32): lanes 0-15 {VGPR0:K=0, VGPR1:K=1}, lanes 16-31 {K=2,K=3}.
// With a = {acc, 0}: A[m][0]=acc[m], A[m][2]=acc[m+16]  =>  D[m][*] = acc[m]+acc[m+16].
// D layout: VGPR v, lanes 0-15 -> row v; lanes 16-31 -> row v+8. In-lane sum of the
// 8 D VGPRs gives rows 0-7 (lanes<16) or rows 8-15 (lanes>=16); one xor-16 shuffle
// completes the exact 32-lane sum in every lane.
__device__ __forceinline__ float wave_sum_wmma(float acc) {
    v2f a; a[0] = acc; a[1] = 0.0f;
    v2f b; b[0] = 1.0f; b[1] = 1.0f;
    v8f c = {};
    v8f d = __builtin_amdgcn_wmma_f32_16x16x4_f32(false, a, false, b,
                                                  (short)0, c, false, false);
    float s = ((d[0] + d[1]) + (d[2] + d[3])) + ((d[4] + d[5]) + (d[6] + d[7]));
    s += __shfl_xor(s, 16, 32);
    return s;
}

// ---------------- K0: init per-column reduction state ----------------
__global__ void k_init(unsigned* __restrict__ tmaxb,
                       unsigned* __restrict__ pminb,
                       unsigned* __restrict__ Mb) {
    int i = blockIdx.x * blockDim.x + threadIdx.x;
    if (i < N_COLS) {
        tmaxb[i] = 0u;          // bits of 0.0f (target >= 0)
        pminb[i] = INF_BITS;    // +inf
    }
    if (i == 0) *Mb = 0u;
}

// ---------------- K1: tmax[n] = max_b target[b,n] ----------------
__global__ void k_tmax(const float* __restrict__ target,
                       unsigned* __restrict__ tmaxb) {
    int n4 = blockIdx.x * TPB + threadIdx.x;
    if (n4 >= N4) return;
    int row0 = blockIdx.y * ROWS_PER_BLOCK;
    float4 m = make_float4(0.f, 0.f, 0.f, 0.f);
    for (int r = 0; r < ROWS_PER_BLOCK; ++r) {
        const float4* tp = (const float4*)(target + (size_t)(row0 + r) * N_COLS);
        if (r + PF_DIST < ROWS_PER_BLOCK)
            __builtin_prefetch((const float4*)(target + (size_t)(row0 + r + PF_DIST) * N_COLS) + n4);
        float4 t = tp[n4];
        m.x = fmaxf(m.x, t.x); m.y = fmaxf(m.y, t.y);
        m.z = fmaxf(m.z, t.z); m.w = fmaxf(m.w, t.w);
    }
    int c = 4 * n4;
    atomicMax(&tmaxb[c + 0], __float_as_uint(m.x));
    atomicMax(&tmaxb[c + 1], __float_as_uint(m.y));
    atomicMax(&tmaxb[c + 2], __float_as_uint(m.z));
    atomicMax(&tmaxb[c + 3], __float_as_uint(m.w));
}

// ---- K2: per-column min of nonzero targets + global masked max (target,loc only) ----
__device__ __forceinline__ void nz_comp(float t, float l, float tm,
                                        float& lmin, float& lM) {
    float loc1 = fminf(fmaxf(l, 0.f), tm);
    if (t > loc1) {
        lmin = fminf(lmin, t);
        lM   = fmaxf(lM, t);
    }
}

__global__ void k_nzstats(const float* __restrict__ target,
                          const float* __restrict__ loc,
                          const unsigned* __restrict__ tmaxb,
                          unsigned* __restrict__ pminb,
                          unsigned* __restrict__ Mb) {
    __shared__ float redM[8];
    int n4 = blockIdx.x * TPB + threadIdx.x;
    bool valid = n4 < N4;
    int row0 = blockIdx.y * ROWS_PER_BLOCK;

    float4 tm = make_float4(0.f, 0.f, 0.f, 0.f);
    if (valid) {
        uint4 tb = ((const uint4*)tmaxb)[n4];
        tm.x = __uint_as_float(tb.x); tm.y = __uint_as_float(tb.y);
        tm.z = __uint_as_float(tb.z); tm.w = __uint_as_float(tb.w);
    }
    float lM = 0.f;
    float4 lmin = make_float4(INFINITY, INFINITY, INFINITY, INFINITY);

    if (valid) {
        for (int r = 0; r < ROWS_PER_BLOCK; ++r) {
            size_t rowoff = (size_t)(row0 + r) * N_COLS;
            if (r + PF_DIST < ROWS_PER_BLOCK) {
                size_t pfo = (size_t)(row0 + r + PF_DIST) * N_COLS;
                __builtin_prefetch((const float4*)(target + pfo) + n4);
                __builtin_prefetch((const float4*)(loc    + pfo) + n4);
            }
            float4 t4 = ((const float4*)(target + rowoff))[n4];
            float4 l4 = ((const float4*)(loc    + rowoff))[n4];
            nz_comp(t4.x, l4.x, tm.x, lmin.x, lM);
            nz_comp(t4.y, l4.y, tm.y, lmin.y, lM);
            nz_comp(t4.z, l4.z, tm.z, lmin.z, lM);
            nz_comp(t4.w, l4.w, tm.w, lmin.w, lM);
        }
        int c = 4 * n4;
        atomicMin(&pminb[c + 0], __float_as_uint(lmin.x));
        atomicMin(&pminb[c + 1], __float_as_uint(lmin.y));
        atomicMin(&pminb[c + 2], __float_as_uint(lmin.z));
        atomicMin(&pminb[c + 3], __float_as_uint(lmin.w));
    }

    // block max-reduce for M, one idempotent atomic per block
    int lane = threadIdx.x & 31, wave = threadIdx.x >> 5;
    float m = lM;
    for (int off = 16; off > 0; off >>= 1)
        m = fmaxf(m, __shfl_xor(m, off, 32));
    if (lane == 0) redM[wave] = m;
    __syncthreads();
    if (threadIdx.x == 0) {
        float tmx = 0.f;
        for (int i = 0; i < 8; ++i) tmx = fmaxf(tmx, redM[i]);
        atomicMax(Mb, __float_as_uint(tmx));
    }
}

// ---------------- K3: fused spike + slab log-likelihood sum ----------------
// Fast-path math: v_rcp_f32 / v_rsq_f32 hardware approximations (~1 ulp), and
// merged logarithms: log(1-q)-log(loc1) = log((1-q)*rcp(loc1));
// log(q)-0.5*log(s2) = log(q*rsq(s2)).
__device__ __forceinline__ void loss_comp(float t, float muv, float s2v, float l, float qv,
                                          float tm, float nzb, float& acc) {
    float loc1 = fminf(fmaxf(l, 0.f), tm);
    if (t <= loc1) {
        // spike: log((1-q)/loc1)   (rcp(0)=+inf keeps the log(0) -> +inf behavior)
        acc += __logf((1.f - qv) * __builtin_amdgcn_rcpf(loc1));
    } else {
        // slab: zero-inflated lognormal on x = t - loc2
        float loc2 = fminf(loc1, nzb);
        float lx = __logf(t - loc2);
        float dd = lx - muv;
        acc += __logf(qv * __builtin_amdgcn_rsqf(s2v))
             - 0.5f * LOG2PI_F
             - dd * dd * (0.5f * __builtin_amdgcn_rcpf(s2v))
             - lx;
    }
}

__global__ void k_loss(const float* __restrict__ target,
                       const float* __restrict__ mu,
                       const float* __restrict__ sigma2,
                       const float* __restrict__ loc,
                       const float* __restrict__ q,
                       const unsigned* __restrict__ tmaxb,
                       const unsigned* __restrict__ pminb,
                       const unsigned* __restrict__ Mb,
                       float* __restrict__ lossP) {
    __shared__ float red8[8];
    int n4 = blockIdx.x * TPB + threadIdx.x;
    bool valid = n4 < N4;
    int row0 = blockIdx.y * ROWS_PER_BLOCK;

    float Mv = __uint_as_float(*Mb);
    float4 tm = make_float4(0.f, 0.f, 0.f, 0.f);
    float4 nzb = make_float4(0.f, 0.f, 0.f, 0.f);
    if (valid) {
        uint4 tb = ((const uint4*)tmaxb)[n4];
        tm.x = __uint_as_float(tb.x); tm.y = __uint_as_float(tb.y);
        tm.z = __uint_as_float(tb.z); tm.w = __uint_as_float(tb.w);
        uint4 pb = ((const uint4*)pminb)[n4];
        nzb.x = 0.999f * ((pb.x == INF_BITS) ? Mv : __uint_as_float(pb.x));
        nzb.y = 0.999f * ((pb.y == INF_BITS) ? Mv : __uint_as_float(pb.y));
        nzb.z = 0.999f * ((pb.z == INF_BITS) ? Mv : __uint_as_float(pb.z));
        nzb.w = 0.999f * ((pb.w == INF_BITS) ? Mv : __uint_as_float(pb.w));
    }

    float acc = 0.f;
    if (valid) {
        for (int r = 0; r < ROWS_PER_BLOCK; ++r) {
            size_t rowoff = (size_t)(row0 + r) * N_COLS;
            if (r + PF_DIST < ROWS_PER_BLOCK) {
                size_t pfo = (size_t)(row0 + r + PF_DIST) * N_COLS;
                __builtin_prefetch((const float4*)(target + pfo) + n4);
                __builtin_prefetch((const float4*)(mu     + pfo) + n4);
                __builtin_prefetch((const float4*)(sigma2 + pfo) + n4);
                __builtin_prefetch((const float4*)(loc    + pfo) + n4);
                __builtin_prefetch((const float4*)(q      + pfo) + n4);
            }
            float4 t4 = ((const float4*)(target + rowoff))[n4];
            float4 m4 = ((const float4*)(mu     + rowoff))[n4];
            float4 s4 = ((const float4*)(sigma2 + rowoff))[n4];
            float4 l4 = ((const float4*)(loc    + rowoff))[n4];
            float4 q4 = ((const float4*)(q      + rowoff))[n4];
            loss_comp(t4.x, m4.x, s4.x, l4.x, q4.x, tm.x, nzb.x, acc);
            loss_comp(t4.y, m4.y, s4.y, l4.y, q4.y, tm.y, nzb.y, acc);
            loss_comp(t4.z, m4.z, s4.z, l4.z, q4.z, tm.z, nzb.z, acc);
            loss_comp(t4.w, m4.w, s4.w, l4.w, q4.w, tm.w, nzb.w, acc);
        }
    }

    int lane = threadIdx.x & 31, wave = threadIdx.x >> 5;
    float w = wave_sum_wmma(acc);
    if (lane == 0) red8[wave] = w;
    __syncthreads();
    if (threadIdx.x == 0) {
        float ts = 0.f;
        for (int i = 0; i < 8; ++i) ts += red8[i];
        lossP[blockIdx.y * gridDim.x + blockIdx.x] = ts;
    }
}

// ---------------- K4: deterministic fixed-order final sum ----------------
__global__ void k_final(const float* __restrict__ lossP,
                        float* __restrict__ out) {
    __shared__ float sm[TPB];
    int t = threadIdx.x;
    float acc = (lossP[t] + lossP[t + 256]) + (lossP[t + 512] + lossP[t + 768]);
    sm[t] = acc;
    __syncthreads();
    for (int s = 128; s > 0; s >>= 1) {
        if (t < s) sm[t] += sm[t + s];
        __syncthreads();
    }
    if (t == 0) out[0] = -sm[0];
}

extern "C" void kernel_launch(void* const* d_in, const int* in_sizes, int n_in,
                              void* d_out, int out_size, void* d_ws, size_t ws_size,
                              hipStream_t stream) {
    const float* target = (const float*)d_in[0];
    const float* mu     = (const float*)d_in[1];
    const float* sigma2 = (const float*)d_in[2];
    const float* loc    = (const float*)d_in[3];
    const float* q      = (const float*)d_in[4];
    float* out = (float*)d_out;

    // workspace layout (words): tmax bits [N], permin bits [N], M bits [1], pad,
    // combined loss partials [NPART]
    unsigned* tmaxb  = (unsigned*)d_ws;
    unsigned* pminb  = tmaxb + N_COLS;          // byte off 32000, 16B aligned
    unsigned* Mb     = pminb + N_COLS;
    float*    lossP  = (float*)(Mb + 16);       // pad past Mb

    dim3 grid(COL_BLOCKS, ROW_BLOCKS);

    k_init   <<<(N_COLS + TPB - 1) / TPB, TPB, 0, stream>>>(tmaxb, pminb, Mb);
    k_tmax   <<<grid, TPB, 0, stream>>>(target, tmaxb);
    k_nzstats<<<grid, TPB, 0, stream>>>(target, loc, tmaxb, pminb, Mb);
    k_loss   <<<grid, TPB, 0, stream>>>(target, mu, sigma2, loc, q, tmaxb, pminb, Mb, lossP);
    k_final  <<<1, TPB, 0, stream>>>(lossP, out);
}